// GCN_11278584119619
// MI455X (gfx1250) — compile-verified
//
#include <hip/hip_runtime.h>
#include <hip/hip_bf16.h>

// GCN forward for MI455X (gfx1250, wave32).
// N=50000 nodes, E=1.6M edges, H=256, OUT=64, 3 tied hidden layers + output.
//
// Per-layer pipeline: y = (h * norm_src) @ W  (fp32 WMMA 16x16x4)
//                     agg = scatter_add(y[src] -> dst)  (L2-resident fp32 atomics)
//                     h' = relu(agg * norm_dst + b)
//
// Working set per layer (~103 MB) fits in the 192 MB L2, so the edge
// aggregation runs at L2 bandwidth rather than HBM.

#define H_FEATS 256
#define OUT_FEATS 64

typedef __attribute__((ext_vector_type(2))) float v2f;
typedef __attribute__((ext_vector_type(8))) float v8f;

// ---------------------------------------------------------------------------
// Zero a float buffer.
__global__ __launch_bounds__(256) void gcn_zero_kernel(float* __restrict__ p, int n) {
    int i = blockIdx.x * 256 + threadIdx.x;
    if (i < n) p[i] = 0.0f;
}

// ---------------------------------------------------------------------------
// Degree accumulation: deg_src[src[e]] += 1, deg_dst[dst[e]] += 1.
__global__ __launch_bounds__(256) void gcn_degree_kernel(const int* __restrict__ src,
                                                         const int* __restrict__ dst,
                                                         float* __restrict__ deg_src,
                                                         float* __restrict__ deg_dst,
                                                         int nedges) {
    int e = blockIdx.x * 256 + threadIdx.x;
    if (e < nedges) {
        unsafeAtomicAdd(&deg_src[src[e]], 1.0f);
        unsafeAtomicAdd(&deg_dst[dst[e]], 1.0f);
    }
}

// In-place: d[i] = rsqrt(max(d[i], 1))  (norm='both')
__global__ __launch_bounds__(256) void gcn_norm_kernel(float* __restrict__ d, int n) {
    int i = blockIdx.x * 256 + threadIdx.x;
    if (i < n) d[i] = __frsqrt_rn(fmaxf(d[i], 1.0f));
}

// ---------------------------------------------------------------------------
// y[nrows x WIDTH] = (x * ns[:,None]) @ w[256 x WIDTH], fp32 WMMA 16x16x4.
// One wave per 16x16 output tile; K=256 -> 64 chained v_wmma_f32_16x16x4_f32.
//
// Fragment layouts per CDNA5 ISA 7.12.2:
//   A 16x4:  lane<16: rows m=l16, v0=K k+0, v1=K k+1 ; lane>=16: K k+2, k+3
//   B 4x16:  lane<16: col n=l16, v0=row k+0, v1=row k+1 ; lane>=16: k+2, k+3
//   D 16x16: VGPR r -> row r (lanes 0-15) / row r+8 (lanes 16-31), col l16
template <int WIDTH>
__global__ __launch_bounds__(256) void gcn_gemm_wmma_kernel(const float* __restrict__ x,
                                                            const float* __restrict__ w,
                                                            const float* __restrict__ ns,
                                                            float* __restrict__ y,
                                                            int nrows) {
    const int tilesN = WIDTH / 16;
    const int lane = threadIdx.x & 31;
    const int tile = blockIdx.x * 8 + (threadIdx.x >> 5);   // 8 waves / block
    const int totalTiles = (nrows / 16) * tilesN;
    if (tile >= totalTiles) return;

    const int tileM = tile / tilesN;
    const int tileN = tile % tilesN;
    const int half = lane >> 4;     // 0: lanes 0-15, 1: lanes 16-31
    const int l16  = lane & 15;

    const int rowA = tileM * 16 + l16;          // A-matrix row held by this lane
    const int colB = tileN * 16 + l16;          // B-matrix col held by this lane
    const float scale = ns[rowA];               // fuse norm_src row scaling into A load
    const float* __restrict__ xrow = x + (size_t)rowA * H_FEATS;

    v8f acc = {};
#pragma unroll 4
    for (int k = 0; k < H_FEATS; k += 4) {
        const int ka = k + half * 2;
        v2f a, b;
        a.x = xrow[ka] * scale;
        a.y = xrow[ka + 1] * scale;
        b.x = w[(size_t)ka * WIDTH + colB];
        b.y = w[(size_t)(ka + 1) * WIDTH + colB];
        // 8 args: (neg_a, A, neg_b, B, c_mod, C, reuse_a, reuse_b)
        acc = __builtin_amdgcn_wmma_f32_16x16x4_f32(false, a, false, b, (short)0, acc,
                                                    false, false);
    }

#pragma unroll
    for (int r = 0; r < 8; ++r) {
        const int row = tileM * 16 + r + half * 8;
        y[(size_t)row * WIDTH + tileN * 16 + l16] = acc[r];
    }
}

// ---------------------------------------------------------------------------
// Edge scatter: agg[dst[e]][:] += y[src[e]][:]. One wave per edge; lane-strided
// coalesced loads + hardware global_atomic_add_f32 (both L2-resident).
template <int WIDTH>
__global__ __launch_bounds__(256) void gcn_scatter_kernel(const float* __restrict__ y,
                                                          const int* __restrict__ src,
                                                          const int* __restrict__ dst,
                                                          float* __restrict__ agg,
                                                          int nedges) {
    const int lane = threadIdx.x & 31;
    const int e = blockIdx.x * 8 + (threadIdx.x >> 5);
    if (e >= nedges) return;
    const float* __restrict__ yr = y + (size_t)src[e] * WIDTH;
    float* __restrict__ ar = agg + (size_t)dst[e] * WIDTH;
#pragma unroll
    for (int j = lane; j < WIDTH; j += 32)
        unsafeAtomicAdd(&ar[j], yr[j]);
}

// ---------------------------------------------------------------------------
// out[i][j] = (relu?)(agg[i][j] * nd[i] + bias[j])   (elementwise; in-place ok)
template <int WIDTH, bool RELU>
__global__ __launch_bounds__(256) void gcn_finish_kernel(const float* __restrict__ agg,
                                                         const float* __restrict__ nd,
                                                         const float* __restrict__ bias,
                                                         float* __restrict__ out,
                                                         int nrows) {
    int idx = blockIdx.x * 256 + threadIdx.x;
    if (idx >= nrows * WIDTH) return;
    int i = idx / WIDTH;
    int j = idx % WIDTH;
    float v = agg[idx] * nd[i] + bias[j];
    out[idx] = RELU ? fmaxf(v, 0.0f) : v;
}

// ---------------------------------------------------------------------------
extern "C" void kernel_launch(void* const* d_in, const int* in_sizes, int n_in,
                              void* d_out, int out_size, void* d_ws, size_t ws_size,
                              hipStream_t stream) {
    const float* feat = (const float*)d_in[0];   // [N, 256] (= eye)
    const float* W1   = (const float*)d_in[1];   // [256, 256]
    const float* b1   = (const float*)d_in[2];   // [256]
    const float* W2   = (const float*)d_in[3];   // [256, 64]
    const float* b2   = (const float*)d_in[4];   // [64]
    const int*   src  = (const int*)d_in[5];     // [E]
    const int*   dst  = (const int*)d_in[6];     // [E]
    float* out = (float*)d_out;

    const int N = in_sizes[0] / H_FEATS;         // 50000 (multiple of 16)
    const int E = in_sizes[5];                   // 1600000

    // Workspace layout (floats): ns[N] | nd[N] | buf0[N*H] | buf1[N*H]
    float* ns   = (float*)d_ws;
    float* nd   = ns + N;
    float* buf0 = nd + N;                        // GEMM output y
    float* buf1 = buf0 + (size_t)N * H_FEATS;    // agg / h

    const int ZB = 256;
    // ---- degree norms ----
    gcn_zero_kernel<<<(2 * N + ZB - 1) / ZB, ZB, 0, stream>>>(ns, 2 * N);
    gcn_degree_kernel<<<(E + ZB - 1) / ZB, ZB, 0, stream>>>(src, dst, ns, nd, E);
    gcn_norm_kernel<<<(2 * N + ZB - 1) / ZB, ZB, 0, stream>>>(ns, 2 * N);

    const int tilesH  = (N / 16) * (H_FEATS / 16);     // waves for H-wide GEMM
    const int tilesO  = (N / 16) * (OUT_FEATS / 16);   // waves for OUT-wide GEMM
    const int gemmH_blocks = (tilesH + 7) / 8;
    const int gemmO_blocks = (tilesO + 7) / 8;
    const int scat_blocks  = (E + 7) / 8;
    const int nhElems = N * H_FEATS;
    const int noElems = N * OUT_FEATS;

    // ---- 3 tied hidden layers ----
    const float* h = feat;
    for (int l = 0; l < 3; ++l) {
        gcn_gemm_wmma_kernel<H_FEATS><<<gemmH_blocks, 256, 0, stream>>>(h, W1, ns, buf0, N);
        gcn_zero_kernel<<<(nhElems + ZB - 1) / ZB, ZB, 0, stream>>>(buf1, nhElems);
        gcn_scatter_kernel<H_FEATS><<<scat_blocks, 256, 0, stream>>>(buf0, src, dst, buf1, E);
        gcn_finish_kernel<H_FEATS, true><<<(nhElems + ZB - 1) / ZB, ZB, 0, stream>>>(
            buf1, nd, b1, buf1, N);
        h = buf1;  // buf1 is consumed by the next GEMM before it is re-zeroed
    }

    // ---- output layer (OUT=64), result straight to d_out ----
    gcn_gemm_wmma_kernel<OUT_FEATS><<<gemmO_blocks, 256, 0, stream>>>(h, W2, ns, buf0, N);
    gcn_zero_kernel<<<(noElems + ZB - 1) / ZB, ZB, 0, stream>>>(buf1, noElems);
    gcn_scatter_kernel<OUT_FEATS><<<scat_blocks, 256, 0, stream>>>(buf0, src, dst, buf1, E);
    gcn_finish_kernel<OUT_FEATS, false><<<(noElems + ZB - 1) / ZB, ZB, 0, stream>>>(
        buf1, nd, b2, out, N);
}